// GraphMessagePassing_13666585936108
// MI455X (gfx1250) — compile-verified
//
#include <hip/hip_runtime.h>
#include <stdint.h>

// Problem constants (from reference): N=100000 nodes, E=1e6 edges, DIM=64.
#define DIM 64

typedef float v2f __attribute__((ext_vector_type(2)));
typedef float v8f __attribute__((ext_vector_type(8)));

// ---------------------------------------------------------------------------
// Kernel 1: zero agg (aliased on d_out, N*64 f32) and deg (d_ws, N f32).
// ---------------------------------------------------------------------------
__global__ void gmp_zero_kernel(float* __restrict__ agg, float* __restrict__ deg,
                                int n_agg, int n_deg) {
  int stride = gridDim.x * blockDim.x;
  for (int k = blockIdx.x * blockDim.x + threadIdx.x; k < n_agg; k += stride)
    agg[k] = 0.0f;
  for (int k = blockIdx.x * blockDim.x + threadIdx.x; k < n_deg; k += stride)
    deg[k] = 0.0f;
}

// ---------------------------------------------------------------------------
// Kernel 2: edge scatter.  agg[i] += x[j], deg[i] += 1.
// 16 lanes per edge; each lane moves a float4 (4 consecutive dims).
// x (25.6MB) and agg (25.6MB) are both L2-resident on MI455X (192MB L2),
// so the random gather + f32 atomics stay on-chip. This phase dominates the
// roofline (~256MB gather + ~256MB atomic traffic, all L2-local).
// ---------------------------------------------------------------------------
__global__ void gmp_scatter_kernel(const float* __restrict__ x,
                                   const long long* __restrict__ ei, // (2,E) int64
                                   float* __restrict__ agg,
                                   float* __restrict__ deg,
                                   int E) {
  int gid = blockIdx.x * blockDim.x + threadIdx.x;
  int e = gid >> 4;        // edge id
  int l = gid & 15;        // 16 lanes per edge, 4 floats each
  if (e >= E) return;
  int i = (int)ei[e];                         // target
  int j = (int)ei[(long long)E + e];          // source
  const float4 v = *(const float4*)(x + (long long)j * DIM + l * 4);
  float* dst = agg + (long long)i * DIM + l * 4;
  atomicAdd(dst + 0, v.x);
  atomicAdd(dst + 1, v.y);
  atomicAdd(dst + 2, v.z);
  atomicAdd(dst + 3, v.w);
  if (l == 0) atomicAdd(deg + i, 1.0f);
}

// ---------------------------------------------------------------------------
// Kernel 3: out = (agg @ W^T + deg*b) / max(deg,1), in place on d_out.
// FP32 WMMA: V_WMMA_F32_16X16X4_F32, one 16x64 output tile per wave.
//   A (16x4 f32):  lane L -> M = L%16; VGPR v holds K = kbase + 2*(L/16) + v
//                  -> contiguous float2 load from the agg row.
//   B (4x16 f32):  lane L -> N = L%16; VGPR v holds K = kbase + 2*(L/16) + v
//                  -> LDS holds K-pair-interleaved W^T:
//                     Bp[kp*128 + 2n + v] = B[2*kp+v][n] = W[n][2*kp+v]
//                     so one aligned ds_load_b64 yields a full B fragment
//                     (no register repacking movs).
//   C/D (16x16):   lane L -> N = L%16; VGPR r holds M = r + 8*(L/16).
// Row indices are clamped (never EXEC-masked) so EXEC==all-ones holds for WMMA.
// ---------------------------------------------------------------------------
__global__ void __launch_bounds__(256) gmp_wmma_kernel(
    const float* __restrict__ W,     // (64,64) row-major: W[n*64+k]
    const float* __restrict__ bias,  // (64,)
    const float* __restrict__ deg,   // (N,)
    float* __restrict__ out,         // in: agg rows; out: final rows (in place)
    int n_nodes, int n_tiles) {
  __shared__ float Bp[DIM * DIM];  // Bp[kp*128 + 2n + v] = W[n][2*kp+v]
  __shared__ float bsh[DIM];

  int tid = threadIdx.x;
  // Cooperative load of W into LDS in K-pair-interleaved B layout.
  for (int idx = tid; idx < DIM * DIM; idx += 256) {
    int n = idx >> 6;        // W row   (output dim)
    int k = idx & 63;        // W col   (input dim / GEMM K)
    Bp[(k >> 1) * 128 + 2 * n + (k & 1)] = W[idx];
  }
  if (tid < DIM) bsh[tid] = bias[tid];
  __syncthreads();

  int wave = tid >> 5;
  int lane = tid & 31;
  int tile = blockIdx.x * 8 + wave;  // 8 waves per block
  if (tile >= n_tiles) return;       // wave-uniform exit

  int row0 = tile * 16;
  int m  = lane & 15;   // A-matrix row within tile
  int kh = lane >> 4;   // K half-select
  int n  = lane & 15;   // B/C/D column within 16-wide tile

  // Clamp A-row so partial tiles read valid memory (each out row depends only
  // on its own agg row, so garbage rows never contaminate valid ones).
  int arow_idx = row0 + m;
  if (arow_idx >= n_nodes) arow_idx = n_nodes - 1;
  const float* arow = out + (long long)arow_idx * DIM;

  v8f acc[4];
  const v8f zero8 = {0.f, 0.f, 0.f, 0.f, 0.f, 0.f, 0.f, 0.f};
#pragma unroll
  for (int nt = 0; nt < 4; ++nt) acc[nt] = zero8;

#pragma unroll
  for (int ks = 0; ks < 16; ++ks) {  // K = 64 in steps of 4
    int k = ks * 4 + 2 * kh;
    v2f a = *(const v2f*)(arow + k);        // A[m][k], A[m][k+1]
    int kp = 2 * ks + kh;                   // K-pair index in LDS layout
    const float* bprow = &Bp[kp * 128];
#pragma unroll
    for (int nt = 0; nt < 4; ++nt) {
      v2f b = *(const v2f*)(bprow + 2 * (nt * 16 + n));  // one ds_load_b64
      acc[nt] = __builtin_amdgcn_wmma_f32_16x16x4_f32(
          /*neg_a=*/false, a, /*neg_b=*/false, b,
          /*c_mod=*/(short)0, acc[nt], /*reuse_a=*/false, /*reuse_b=*/false);
    }
  }

  // Epilogue: out[m][n] = (acc + deg*b) / max(deg,1)
#pragma unroll
  for (int r = 0; r < 8; ++r) {
    int mg = row0 + r + 8 * kh;
    int mg_c = (mg < n_nodes) ? mg : (n_nodes - 1);
    float d = deg[mg_c];
    float inv = 1.0f / fmaxf(d, 1.0f);
#pragma unroll
    for (int nt = 0; nt < 4; ++nt) {
      float val = (acc[nt][r] + d * bsh[nt * 16 + n]) * inv;
      if (mg < n_nodes) out[(long long)mg * DIM + nt * 16 + n] = val;
    }
  }
}

// ---------------------------------------------------------------------------
extern "C" void kernel_launch(void* const* d_in, const int* in_sizes, int n_in,
                              void* d_out, int out_size, void* d_ws, size_t ws_size,
                              hipStream_t stream) {
  const float*      x  = (const float*)d_in[0];      // (N,64) f32
  const long long*  ei = (const long long*)d_in[1];  // (2,E) i64
  const float*      W  = (const float*)d_in[2];      // (64,64) f32
  const float*      b  = (const float*)d_in[3];      // (64,) f32
  float* out = (float*)d_out;                        // (N,64) f32, also agg
  float* deg = (float*)d_ws;                         // N f32 scratch

  int n_nodes = in_sizes[0] / DIM;
  int E = in_sizes[1] / 2;

  // 1) zero agg (= d_out) and deg
  gmp_zero_kernel<<<512, 256, 0, stream>>>(out, deg, n_nodes * DIM, n_nodes);

  // 2) scatter-add x[j] into agg[i]; count degrees
  long long threads = (long long)E * 16;
  int blocks = (int)((threads + 255) / 256);
  gmp_scatter_kernel<<<blocks, 256, 0, stream>>>(x, ei, out, deg, E);

  // 3) fused WMMA GEMM + bias + degree-normalize, in place on d_out
  int n_tiles = (n_nodes + 15) / 16;
  int gblocks = (n_tiles + 7) / 8;
  gmp_wmma_kernel<<<gblocks, 256, 0, stream>>>(W, b, deg, out, n_nodes, n_tiles);
}